// GraphProp_27582279975441
// MI455X (gfx1250) — compile-verified
//
#include <hip/hip_runtime.h>
#include <math.h>

#define Hdim 128

typedef float v2f __attribute__((ext_vector_type(2)));
typedef float v8f __attribute__((ext_vector_type(8)));
typedef unsigned int u32x4 __attribute__((ext_vector_type(4)));
typedef unsigned int u32x8 __attribute__((ext_vector_type(8)));

__device__ __forceinline__ v8f wmma4(v2f a, v2f b, v8f c) {
  // D = A(16x4 f32) x B(4x16 f32) + C(16x16 f32), exact fp32 matrix op
  return __builtin_amdgcn_wmma_f32_16x16x4_f32(false, a, false, b, (short)0, c,
                                               false, false);
}

// ---- Tensor Data Mover: 2D tile (rows x 128 f32) global -> LDS ----
// tensor_dim1 = rows_avail gives hardware zero-fill past the last node.
// pad_interval=6 (128 DWORDs) + pad_amount=3 (4 DWORDs) recreates the
// bank-conflict-free 132-float LDS row stride inside the DMA engine.
__device__ __forceinline__ void tdm_load_tile(const float* gptr,
                                              unsigned lds_byte_off,
                                              unsigned rows_avail) {
  const unsigned long long ga = (unsigned long long)(uintptr_t)gptr;
  u32x4 g0;
  g0[0] = 1u;                                   // count=1 (valid), no gather
  g0[1] = lds_byte_off;                         // lds_addr
  g0[2] = (unsigned)(ga & 0xffffffffu);         // global_addr[31:0]
  g0[3] = (unsigned)((ga >> 32) & 0x01ffffffu)  // global_addr[56:32]
          | (2u << 30);                         // type=2 ("image")
  u32x8 g1;
  g1[0] = (2u << 16)      // data_size = 4 bytes
          | (1u << 20)    // pad_enable
          | (6u << 22)    // pad_interval: 128 DWORDs
          | (3u << 25);   // pad_amount: 4 DWORDs
  g1[1] = (128u << 16);                    // tensor_dim0 = 128 (lo16)
  g1[2] = (rows_avail & 0xffffu) << 16;    // dim0 hi=0 | tensor_dim1 lo16
  g1[3] = (rows_avail >> 16)               // tensor_dim1 hi16
          | (128u << 16);                  // tile_dim0 = 128
  g1[4] = 16u;                             // tile_dim1 = 16, tile_dim2 = 0
  g1[5] = 128u;                            // tensor_dim0_stride = 128
  g1[6] = 0u;
  g1[7] = 0u;
  asm volatile("tensor_load_to_lds %0, %1" ::"s"(g0), "s"(g1) : "memory");
}

// ---------------- CSR build (by dst) ----------------
__global__ void k_hist(const int* __restrict__ dst, int* __restrict__ cnt, int E) {
  int i = blockIdx.x * blockDim.x + threadIdx.x;
  if (i < E) atomicAdd(&cnt[dst[i]], 1);
}

__global__ __launch_bounds__(1024) void k_scan(int* __restrict__ cnt,
                                               int* __restrict__ rowptr, int n) {
  __shared__ int sdata[1024];
  __shared__ int s_carry;
  const int tid = threadIdx.x;
  if (tid == 0) s_carry = 0;
  __syncthreads();
  for (int base = 0; base < n; base += 1024) {
    const int c = s_carry;
    const int i = base + tid;
    int v = (i < n) ? cnt[i] : 0;
    sdata[tid] = v;
    __syncthreads();
    for (int off = 1; off < 1024; off <<= 1) {
      int t = (tid >= off) ? sdata[tid - off] : 0;
      __syncthreads();
      sdata[tid] += t;
      __syncthreads();
    }
    if (i < n) {
      int excl = c + sdata[tid] - v;
      rowptr[i] = excl;
      cnt[i] = excl;  // doubles as scatter cursor
    }
    int tot = sdata[1023];
    __syncthreads();
    if (tid == 0) s_carry = c + tot;
    __syncthreads();
  }
  if (tid == 0) rowptr[n] = s_carry;
}

__global__ void k_scatter(const int* __restrict__ dst, int* __restrict__ cursor,
                          int* __restrict__ eid, int E) {
  int i = blockIdx.x * blockDim.x + threadIdx.x;
  if (i < E) {
    int p = atomicAdd(&cursor[dst[i]], 1);
    eid[p] = i;
  }
}

// ------------- per-node mailbox reduction (no atomics) -------------
template <int WITH_E>
__global__ __launch_bounds__(256) void k_agg(
    const float* __restrict__ hcur, const float* __restrict__ e,
    const int* __restrict__ src, const int* __restrict__ rowptr,
    const int* __restrict__ eid, float* __restrict__ S, float* __restrict__ Es,
    int n) {
  const int lane = threadIdx.x & 31;
  const int gw = (blockIdx.x * blockDim.x + threadIdx.x) >> 5;
  const int nw = (gridDim.x * blockDim.x) >> 5;
  for (int v = gw; v < n; v += nw) {
    const int beg = rowptr[v], end = rowptr[v + 1];
    float4 aH = make_float4(0.f, 0.f, 0.f, 0.f);
    float4 aE = make_float4(0.f, 0.f, 0.f, 0.f);
    for (int j = beg; j < end; ++j) {
      const int ei = eid[j];
      const int s = src[ei];
      if (WITH_E && (j + 1 < end)) {
        __builtin_prefetch(e + (size_t)eid[j + 1] * Hdim, 0, 1);
      }
      float4 hv = ((const float4*)(hcur + (size_t)s * Hdim))[lane];
      aH.x += hv.x; aH.y += hv.y; aH.z += hv.z; aH.w += hv.w;
      if (WITH_E) {
        float4 ev = ((const float4*)(e + (size_t)ei * Hdim))[lane];
        aE.x += ev.x; aE.y += ev.y; aE.z += ev.z; aE.w += ev.w;
      }
    }
    ((float4*)(S + (size_t)v * Hdim))[lane] = aH;
    if (WITH_E) ((float4*)(Es + (size_t)v * Hdim))[lane] = aE;
  }
}

// ------------- fused dense layer: a = f(h,S,Es,deg); h' = GRU(a, h) -------------
// Block = 256 threads = 8 waves; one 16-node M-tile per block.
__global__ __launch_bounds__(256) void k_dense(
    const float* __restrict__ h_in, const float* __restrict__ S,
    const float* __restrict__ Es, const int* __restrict__ rowptr,
    const float* __restrict__ Wm,   // [256,384] row-major
    const float* __restrict__ bm,   // [256]
    const float* __restrict__ Wih,  // [384,256]
    const float* __restrict__ Whh,  // [384,128]
    const float* __restrict__ bih,  // [384]
    const float* __restrict__ bhh,  // [384]
    float* __restrict__ h_out, int n) {
  __shared__ float sh_h[16][132];  // 132 = 128 + TDM pad (4 DW) -> no bank conflicts
  __shared__ float sh_S[16][132];
  __shared__ float sh_E[16][132];
  __shared__ float sh_a[16][260];
  __shared__ float sh_pres[16];
  __shared__ float sh_inv[16];

  const int tid = threadIdx.x;
  const int lane = tid & 31;
  const int wave = tid >> 5;
  const int m0 = blockIdx.x * 16;

  // ---- TDM stages all three 16x128 tiles; hardware zero-fills the tail ----
  if (wave == 0) {
    const unsigned rows = (unsigned)(n - m0);
    tdm_load_tile(h_in + (size_t)m0 * Hdim,
                  (unsigned)(uintptr_t)&sh_h[0][0], rows);
    tdm_load_tile(S + (size_t)m0 * Hdim,
                  (unsigned)(uintptr_t)&sh_S[0][0], rows);
    tdm_load_tile(Es + (size_t)m0 * Hdim,
                  (unsigned)(uintptr_t)&sh_E[0][0], rows);
    __builtin_amdgcn_s_wait_tensorcnt(0);
  }
  if (tid < 16) {
    const int node = m0 + tid;
    float deg = 0.f;
    if (node < n) deg = (float)(rowptr[node + 1] - rowptr[node]);
    sh_pres[tid] = deg > 0.f ? 1.f : 0.f;
    sh_inv[tid] = 1.f / fmaxf(deg, 1.f);
  }
  __syncthreads();

  // fragment addressing (wave32 fp32 WMMA layout)
  const int arow = lane & 15;          // A: row of M
  const int akoff = (lane >> 4) << 1;  // A/B: K sub-offset 0 or 2
  const int bn = lane & 15;            // B: column within 16-wide N tile
  const int chalf = (lane >> 4) << 3;  // C/D: +8 row offset for upper lanes
  const int ccol = lane & 15;          // C/D: column
  const v8f Z8 = {0.f, 0.f, 0.f, 0.f, 0.f, 0.f, 0.f, 0.f};

  // ---- stage 1: a[16,256]; wave owns 2 of the 16 column tiles ----
  for (int t = 0; t < 2; ++t) {
    const int nb = (wave * 2 + t) * 16;
    const float* wrow = Wm + (size_t)(nb + bn) * 384;
    v8f acc1 = Z8;  // h @ Wd^T   (dst block, cols 0:128 of Wm)
    v8f acc2 = Z8;  // S @ Ws^T + Es @ We^T
#pragma unroll 8
    for (int k0 = 0; k0 < Hdim; k0 += 4) {
      const int kk = k0 + akoff;
      v2f A, B;
      A.x = sh_h[arow][kk]; A.y = sh_h[arow][kk + 1];
      B.x = wrow[kk];       B.y = wrow[kk + 1];
      acc1 = wmma4(A, B, acc1);
      A.x = sh_S[arow][kk]; A.y = sh_S[arow][kk + 1];
      B.x = wrow[128 + kk]; B.y = wrow[128 + kk + 1];
      acc2 = wmma4(A, B, acc2);
      A.x = sh_E[arow][kk]; A.y = sh_E[arow][kk + 1];
      B.x = wrow[256 + kk]; B.y = wrow[256 + kk + 1];
      acc2 = wmma4(A, B, acc2);
    }
    const float bmv = bm[nb + ccol];
#pragma unroll
    for (int g = 0; g < 8; ++g) {
      const int r = chalf + g;
      sh_a[r][nb + ccol] = sh_pres[r] * (acc1[g] + bmv) + sh_inv[r] * acc2[g];
    }
  }
  __syncthreads();

  // ---- stage 2: GRU; wave owns output channels [wave*16, wave*16+16) ----
  const int jb = wave * 16;
  const float* wih_r = Wih + (size_t)(jb + bn) * 256;
  const float* wih_z = Wih + (size_t)(128 + jb + bn) * 256;
  const float* wih_n = Wih + (size_t)(256 + jb + bn) * 256;
  v8f gir = Z8, giz = Z8, gin = Z8, ghr = Z8, ghz = Z8, ghn = Z8;
#pragma unroll 4
  for (int k0 = 0; k0 < 256; k0 += 4) {  // gi = a @ Wih^T
    const int kk = k0 + akoff;
    v2f A, B;
    A.x = sh_a[arow][kk]; A.y = sh_a[arow][kk + 1];
    B.x = wih_r[kk]; B.y = wih_r[kk + 1]; gir = wmma4(A, B, gir);
    B.x = wih_z[kk]; B.y = wih_z[kk + 1]; giz = wmma4(A, B, giz);
    B.x = wih_n[kk]; B.y = wih_n[kk + 1]; gin = wmma4(A, B, gin);
  }
  const float* whh_r = Whh + (size_t)(jb + bn) * Hdim;
  const float* whh_z = Whh + (size_t)(128 + jb + bn) * Hdim;
  const float* whh_n = Whh + (size_t)(256 + jb + bn) * Hdim;
#pragma unroll 4
  for (int k0 = 0; k0 < Hdim; k0 += 4) {  // gh = h @ Whh^T
    const int kk = k0 + akoff;
    v2f A, B;
    A.x = sh_h[arow][kk]; A.y = sh_h[arow][kk + 1];
    B.x = whh_r[kk]; B.y = whh_r[kk + 1]; ghr = wmma4(A, B, ghr);
    B.x = whh_z[kk]; B.y = whh_z[kk + 1]; ghz = wmma4(A, B, ghz);
    B.x = whh_n[kk]; B.y = whh_n[kk + 1]; ghn = wmma4(A, B, ghn);
  }
  const int j = jb + ccol;
  const float bir = bih[j], biz = bih[128 + j], bin = bih[256 + j];
  const float bhr = bhh[j], bhz = bhh[128 + j], bhn = bhh[256 + j];
#pragma unroll
  for (int g = 0; g < 8; ++g) {
    const int r = chalf + g;
    const int node = m0 + r;
    float rg = 1.f / (1.f + __expf(-(gir[g] + bir + ghr[g] + bhr)));
    float zg = 1.f / (1.f + __expf(-(giz[g] + biz + ghz[g] + bhz)));
    float ng = tanhf(gin[g] + bin + rg * (ghn[g] + bhn));
    float hnew = (1.f - zg) * ng + zg * sh_h[r][j];
    if (node < n) h_out[(size_t)node * Hdim + j] = hnew;
  }
}

// ---------------------------------------------------------------------------
extern "C" void kernel_launch(void* const* d_in, const int* in_sizes, int n_in,
                              void* d_out, int out_size, void* d_ws,
                              size_t ws_size, hipStream_t stream) {
  (void)n_in; (void)out_size; (void)ws_size;
  const float* h = (const float*)d_in[0];
  const float* e = (const float*)d_in[1];
  const int* src = (const int*)d_in[2];
  const int* dst = (const int*)d_in[3];
  const float* Wm[2]  = {(const float*)d_in[4],  (const float*)d_in[10]};
  const float* bm[2]  = {(const float*)d_in[5],  (const float*)d_in[11]};
  const float* Wih[2] = {(const float*)d_in[6],  (const float*)d_in[12]};
  const float* Whh[2] = {(const float*)d_in[7],  (const float*)d_in[13]};
  const float* bih[2] = {(const float*)d_in[8],  (const float*)d_in[14]};
  const float* bhh[2] = {(const float*)d_in[9],  (const float*)d_in[15]};
  const int N_ = in_sizes[0] / Hdim;
  const int E_ = in_sizes[2];

  char* p = (char*)d_ws;
  auto carve = [&](size_t bytes) {
    char* r = p;
    p += (bytes + 255) & ~(size_t)255;
    return r;
  };
  int* rowptr = (int*)carve(sizeof(int) * (size_t)(N_ + 1));
  int* cursor = (int*)carve(sizeof(int) * (size_t)N_);
  int* eid    = (int*)carve(sizeof(int) * (size_t)E_);
  float* Es   = (float*)carve(sizeof(float) * (size_t)N_ * Hdim);
  float* Sbuf = (float*)carve(sizeof(float) * (size_t)N_ * Hdim);
  float* h1   = (float*)carve(sizeof(float) * (size_t)N_ * Hdim);

  // CSR by dst (built once; reused by both layers)
  hipMemsetAsync(cursor, 0, sizeof(int) * (size_t)N_, stream);
  k_hist<<<(E_ + 255) / 256, 256, 0, stream>>>(dst, cursor, E_);
  k_scan<<<1, 1024, 0, stream>>>(cursor, rowptr, N_);
  k_scatter<<<(E_ + 255) / 256, 256, 0, stream>>>(dst, cursor, eid, E_);

  const int aggBlocks = (N_ + 7) / 8;     // 8 waves/block, wave per node
  const int gemmBlocks = (N_ + 15) / 16;  // 16-node M-tile per block

  // layer 0: Es (layer-invariant) + S computed in one pass over edges
  k_agg<1><<<aggBlocks, 256, 0, stream>>>(h, e, src, rowptr, eid, Sbuf, Es, N_);
  k_dense<<<gemmBlocks, 256, 0, stream>>>(h, Sbuf, Es, rowptr, Wm[0], bm[0],
                                          Wih[0], Whh[0], bih[0], bhh[0], h1, N_);
  // layer 1: only S depends on h; e is never re-read
  k_agg<0><<<aggBlocks, 256, 0, stream>>>(h1, e, src, rowptr, eid, Sbuf, Es, N_);
  k_dense<<<gemmBlocks, 256, 0, stream>>>(h1, Sbuf, Es, rowptr, Wm[1], bm[1],
                                          Wih[1], Whh[1], bih[1], bhh[1],
                                          (float*)d_out, N_);
}